// GroupDW_824633721411
// MI455X (gfx1250) — compile-verified
//
#include <hip/hip_runtime.h>
#include <hip/hip_bf16.h>

typedef __attribute__((ext_vector_type(2))) float v2f;
typedef __attribute__((ext_vector_type(8))) float v8f;

#define NBR     3       // three branches (x11,z11) (x12,z12) (x21,z21)
#define KH      7
#define KW      7
#define XH      31
#define XW      31
#define OH      25
#define OW      25
#define XROWS   38      // 31 data rows + 7 zero pad rows (p up to 31, +i up to 6)
#define XPITCH  33      // odd pitch -> conflict-free column-strided A loads
#define ZPITCH  64      // z row padded both sides with zeros
#define ZOFF    32      // z[j] stored at ZOFF+j; index ZOFF+(w-q) always in [1,63]
#define THREADS 128     // 4 waves of 32; each wave owns one 16x16 C quadrant

// Accumulate sum_r sum_i  A(x rows) x T_i(z band) over K-steps [KSLO, KSLO+KSN).
// KSLO/KSN are compile-time so the ks loop fully unrolls with immediate offsets.
template <int KSLO, int KSN>
__device__ __forceinline__ void accum_band(
    const float (*xs)[XROWS][XPITCH], const float (*zp)[KH][ZPITCH],
    int p, int q, int kb, v8f& acc)
{
    for (int r = 0; r < NBR; ++r) {
        for (int i = 0; i < KH; ++i) {
            const float* xrow = &xs[r][p + i][0];
            const float* zrow = &zp[r][i][ZOFF];
#pragma unroll
            for (int ks = KSLO; ks < KSLO + KSN; ++ks) {
                const int w = (ks << 2) + kb;      // x column / Toeplitz row, <= 30
                v2f a, b;
                a.x = xrow[w];
                a.y = xrow[w + 1];
                const int d = w - q;               // band offset, in [-31, 31]
                b.x = zrow[d];                     // zero outside the 7-wide band
                b.y = zrow[d + 1];
                // 8 args: (neg_a, A, neg_b, B, c_mod, C, reuse_a, reuse_b)
                acc = __builtin_amdgcn_wmma_f32_16x16x4_f32(
                    false, a, false, b, (short)0, acc, false, false);
            }
        }
    }
}

__global__ __launch_bounds__(THREADS)
void groupdw_wmma_kernel(const float* __restrict__ z0, const float* __restrict__ z1,
                         const float* __restrict__ z2, const float* __restrict__ x0,
                         const float* __restrict__ x1, const float* __restrict__ x2,
                         const float* __restrict__ wgt, float* __restrict__ out,
                         int nbc)
{
    __shared__ float xs[NBR][XROWS][XPITCH];   // x tiles, zero padded
    __shared__ float zp[NBR][KH][ZPITCH];      // softmax-scaled z rows, zero padded

    const int bc  = blockIdx.x;
    const int tid = threadIdx.x;

    // ---- softmax over the 3 mixing weights (tiny; redundant per thread) ----
    const float w0 = wgt[0], w1 = wgt[1], w2 = wgt[2];
    const float mx = fmaxf(w0, fmaxf(w1, w2));
    const float e0 = __expf(w0 - mx), e1 = __expf(w1 - mx), e2 = __expf(w2 - mx);
    const float inv = 1.0f / (e0 + e1 + e2);
    const float wr0 = e0 * inv, wr1 = e1 * inv, wr2 = e2 * inv;

    // ---- zero LDS (padding must contribute 0 to the WMMA accumulation) ----
    {
        float* xsf = &xs[0][0][0];
        for (int i = tid; i < NBR * XROWS * XPITCH; i += THREADS) xsf[i] = 0.0f;
        float* zpf = &zp[0][0][0];
        for (int i = tid; i < NBR * KH * ZPITCH; i += THREADS) zpf[i] = 0.0f;
    }
    __syncthreads();

    // ---- stage x and (weight-scaled) z into LDS, coalesced ----
    for (int idx = tid; idx < NBR * XH * XW; idx += THREADS) {
        const int r   = idx / (XH * XW);
        const int rem = idx - r * (XH * XW);
        const int row = rem / XW;
        const int col = rem - row * XW;
        const float* src = (r == 0) ? x0 : (r == 1) ? x1 : x2;
        xs[r][row][col] = src[(size_t)bc * (XH * XW) + rem];
    }
    for (int idx = tid; idx < NBR * KH * KW; idx += THREADS) {
        const int r   = idx / (KH * KW);
        const int rem = idx - r * (KH * KW);
        const int i   = rem / KW;
        const int j   = rem - i * KW;
        const float* src = (r == 0) ? z0 : (r == 1) ? z1 : z2;
        const float sw   = (r == 0) ? wr0 : (r == 1) ? wr1 : wr2;
        zp[r][i][ZOFF + j] = src[(size_t)bc * (KH * KW) + rem] * sw;
    }

    // ---- prefetch next channel's x into L2 (global_prefetch_b8) ----
    if (bc + 1 < nbc && tid < 31) {   // 31 lanes x 128B covers a 3844B x tile
        const size_t nb = (size_t)(bc + 1) * (XH * XW) + (size_t)tid * 32;
        __builtin_prefetch(x0 + nb, 0, 1);
        __builtin_prefetch(x1 + nb, 0, 1);
        __builtin_prefetch(x2 + nb, 0, 1);
    }
    __syncthreads();

    // ---- per-wave 16x16 output quadrant via V_WMMA_F32_16X16X4_F32 ----
    // out[25x25] = sum_r sum_i  X_i[25x31] x (wr * T_i)[31x25]
    //   X_i rows i..i+24 of x;  T_i[w,q] = z[i, w-q]  (banded Toeplitz, read
    //   from the zero-padded zp row so no predication is needed).
    const int lane = tid & 31;
    const int wvu  = __builtin_amdgcn_readfirstlane(tid) >> 5;  // uniform wave id
    const int mt   = (wvu >> 1) << 4;        // 0 or 16 (output-row tile)
    const int nt   = (wvu & 1) << 4;         // 0 or 16 (output-col tile)
    const int mr   = lane & 15;
    const int kb   = (lane >> 4) << 1;       // K base for this lane: 0 or 2
    const int p    = mt + mr;                // A-fragment output row (row p+i of x)
    const int q    = nt + mr;                // B-fragment output column

    v8f acc = {0.f, 0.f, 0.f, 0.f, 0.f, 0.f, 0.f, 0.f};

    // Band pruning: T_i[w,q] nonzero only for q <= w <= q+6, so whole K-steps
    // vanish per N-tile.  nt=0 -> w<=21 -> ks 0..5;  nt=16 -> w>=16 -> ks 4..7.
    // nt is wave-uniform, so this is a scalar branch (EXEC stays all-ones).
    if (nt == 0) accum_band<0, 6>(xs, zp, p, q, kb, acc);
    else         accum_band<4, 4>(xs, zp, p, q, kb, acc);

    // ---- store valid region of the C tile ----
    // C layout: VGPR v, lane l -> M = v + 8*(l/16), N = l%16
    const int qn = nt + (lane & 15);
    const int ph = mt + ((lane >> 4) << 3);
    if (qn < OW) {
        float* obase = out + (size_t)bc * (OH * OW) + qn;
#pragma unroll
        for (int v = 0; v < 8; ++v) {
            const int pm = ph + v;
            if (pm < OH) obase[(size_t)pm * OW] = acc[v];
        }
    }
}

extern "C" void kernel_launch(void* const* d_in, const int* in_sizes, int n_in,
                              void* d_out, int out_size, void* d_ws, size_t ws_size,
                              hipStream_t stream) {
    // setup_inputs order: z11, z12, z21, x11, x12, x21, weight
    const float* z0  = (const float*)d_in[0];
    const float* z1  = (const float*)d_in[1];
    const float* z2  = (const float*)d_in[2];
    const float* x0  = (const float*)d_in[3];
    const float* x1  = (const float*)d_in[4];
    const float* x2  = (const float*)d_in[5];
    const float* wgt = (const float*)d_in[6];
    float* out = (float*)d_out;

    const int nbc = in_sizes[0] / (KH * KW);   // B*C = 8192 for the reference shapes

    groupdw_wmma_kernel<<<dim3(nbc), dim3(THREADS), 0, stream>>>(
        z0, z1, z2, x0, x1, x2, wgt, out, nbc);
}